// perpendicularity_42460046688731
// MI455X (gfx1250) — compile-verified
//
#include <hip/hip_runtime.h>
#include <hip/hip_bf16.h>

typedef __attribute__((ext_vector_type(2))) float        v2f;
typedef __attribute__((ext_vector_type(8))) float        v8f;
typedef __attribute__((ext_vector_type(4))) unsigned int v4u;
typedef __attribute__((ext_vector_type(8))) int          v8i;
typedef __attribute__((ext_vector_type(4))) int          v4i;

// Problem constants (from reference: x is [64, 1048576] fp32)
#define N_ROWS   64
#define DIM      1048576
#define KBLOCK   2048              // K columns per block
#define KC       128               // K columns per LDS stage
#define STAGES   (KBLOCK / KC)     // 16
#define NBLOCKS  (DIM / KBLOCK)    // 512
#define LDS_S    (KC + 4)          // row stride in floats; 132 % 64 == 4 -> conflict-free b64 gathers
#define BUF_FLTS (N_ROWS * LDS_S)  // 8448 floats = 33,792 B per buffer
#define BUF_BYTES (BUF_FLTS * 4)
#define NWAVES   10                // upper-triangle 16x16 tiles of a 64x64 gram
#define BLOCKT   (NWAVES * 32)     // 320 threads

__global__ void __launch_bounds__(256)
zero_gram_kernel(float* __restrict__ gram) {
    int t = blockIdx.x * blockDim.x + threadIdx.x;
    if (t < N_ROWS * N_ROWS) gram[t] = 0.0f;
}

// Compiler-visible stand-in for the TDM write: capture the LDS buffer address
// in an asm with a memory clobber (input-only, so the loads themselves keep
// their addrspace(3) provenance and still lower to ds_load, not flat_load).
__device__ __forceinline__ void lds_publish(const float* p) {
    asm volatile("" : : "v"((unsigned long long)(uintptr_t)p) : "memory");
}

// One TDM descriptor moves a 64-row x 128-col fp32 tile (32 KB) global->LDS,
// with HW padding of 4 dwords after every 128 dwords => LDS row stride 132.
__device__ __forceinline__ void tdm_load_stage(const float* gp, unsigned lds_off) {
    const unsigned long long ga = (unsigned long long)(uintptr_t)gp;
    v4u g0;
    g0[0] = 1u;                                           // count=1 (valid), user mode
    g0[1] = lds_off;                                      // lds_addr (bytes)
    g0[2] = (unsigned)(ga & 0xFFFFFFFFu);                 // global_addr[31:0]
    g0[3] = (unsigned)((ga >> 32) & 0x01FFFFFFu)          // global_addr[56:32]
          | (2u << 30);                                   // type=2 ("image")
    v8i g1;
    g1[0] = (int)((2u << 16)                              // data_size: 4 bytes
                | (1u << 20)                              // pad_enable
                | (6u << 22)                              // pad_interval: 128 dwords
                | (3u << 25));                            // pad_amount: 4 dwords
    g1[1] = (int)(((unsigned)DIM & 0xFFFFu) << 16);       // tensor_dim0[15:0]
    g1[2] = (int)((((unsigned)DIM >> 16) & 0xFFFFu)       // tensor_dim0[31:16]
                | ((unsigned)N_ROWS << 16));              // tensor_dim1[15:0]
    g1[3] = (int)((unsigned)KC << 16);                    // tile_dim0 = 128
    g1[4] = (int)(unsigned)N_ROWS;                        // tile_dim1 = 64, tile_dim2 = 0
    g1[5] = (int)(unsigned)DIM;                           // tensor_dim0_stride[31:0]
    g1[6] = 0;                                            // stride0[47:32], stride1 lo
    g1[7] = 0;
    const v4i z4 = {0, 0, 0, 0};
#if defined(__clang_major__) && (__clang_major__ >= 23)
    const v8i z8 = {0, 0, 0, 0, 0, 0, 0, 0};
    __builtin_amdgcn_tensor_load_to_lds(g0, g1, z4, z4, z8, 0);
#else
    __builtin_amdgcn_tensor_load_to_lds(g0, g1, z4, z4, 0);
#endif
}

__global__ void __launch_bounds__(BLOCKT)
gram_wmma_kernel(const float* __restrict__ x, float* __restrict__ gram) {
    __shared__ float lds[2][BUF_FLTS];   // 2 x 33,792 B = 67,584 B (double buffer)

    const int tid    = threadIdx.x;
    const int waveId = tid >> 5;
    const int lane   = tid & 31;
    const int half   = lane >> 4;           // selects K pair {0,1} vs {2,3}
    const int m      = lane & 15;           // row within tile (A: M, B: N)

    // upper-triangle tile assignment (ti <= tj)
    const int TI[NWAVES] = {0,0,0,0,1,1,1,2,2,3};
    const int TJ[NWAVES] = {0,1,2,3,1,2,3,2,3,3};
    const int ti = TI[waveId];
    const int tj = TJ[waveId];

    const int offA = (ti * 16 + m) * LDS_S;
    const int offB = (tj * 16 + m) * LDS_S;

    v8f acc = {0.f, 0.f, 0.f, 0.f, 0.f, 0.f, 0.f, 0.f};

    const int kbase0 = blockIdx.x * KBLOCK;

    // prologue: DMA stage 0 into buffer 0 (wave 0 only; uniform branch)
    if (waveId == 0) tdm_load_stage(x + kbase0, 0u);

    int p = 0;
    for (int s = 0; s < STAGES; ++s) {
        // buffer p ready?  (wave0 has exactly one outstanding TDM op here)
        if (waveId == 0) __builtin_amdgcn_s_wait_tensorcnt(0);
        __syncthreads();

        // overlap: DMA next stage into the other buffer while we compute
        if (waveId == 0 && (s + 1) < STAGES)
            tdm_load_stage(x + kbase0 + (s + 1) * KC, (unsigned)((p ^ 1) * BUF_BYTES));

        // publish the DMA'd buffer to the optimizer (captures address + clobber)
        lds_publish(&lds[p][0]);

        // each wave: 32 x V_WMMA_F32_16X16X4_F32 over this stage
        #pragma unroll 4
        for (int k = 0; k < KC; k += 4) {
            const int kk = k + 2 * half;
            v2f a, b;
            a[0] = lds[p][offA + kk];
            a[1] = lds[p][offA + kk + 1];
            b[0] = lds[p][offB + kk];
            b[1] = lds[p][offB + kk + 1];
            acc = __builtin_amdgcn_wmma_f32_16x16x4_f32(
                false, a, false, b, (short)0, acc, false, false);
        }
        asm volatile("" ::: "memory");   // keep stage-s loads inside stage s
        __syncthreads();                 // all done reading buffer p before refill
        p ^= 1;
    }

    // flush partial tile: C/D layout -> VGPR v holds M = v + 8*half, N = m
    #pragma unroll
    for (int v = 0; v < 8; ++v) {
        const int row = ti * 16 + v + 8 * half;
        const int col = tj * 16 + m;
        unsafeAtomicAdd(&gram[row * N_ROWS + col], acc[v]);
    }
}

__global__ void __launch_bounds__(256)
finalize_kernel(const float* __restrict__ gram, float* __restrict__ out) {
    __shared__ float rn[N_ROWS];
    __shared__ float red[256];
    const int t = threadIdx.x;

    if (t < N_ROWS) {
        const float nrm = sqrtf(gram[t * N_ROWS + t]);
        rn[t] = 1.0f / fmaxf(nrm, 1e-8f);
    }
    __syncthreads();

    float p = 0.0f;
    for (int idx = t; idx < N_ROWS * N_ROWS; idx += 256) {
        const int i = idx >> 6;
        const int j = idx & 63;
        if (j > i) p += gram[idx] * rn[i] * rn[j];
    }
    red[t] = p;
    __syncthreads();
    for (int s = 128; s > 0; s >>= 1) {
        if (t < s) red[t] += red[t + s];
        __syncthreads();
    }
    if (t == 0) out[0] = red[0] / 2016.0f;   // 64*63/2 pairs
}

extern "C" void kernel_launch(void* const* d_in, const int* in_sizes, int n_in,
                              void* d_out, int out_size, void* d_ws, size_t ws_size,
                              hipStream_t stream) {
    const float* x = (const float*)d_in[0];
    float* gram    = (float*)d_ws;          // 64*64*4 = 16 KB scratch
    float* out     = (float*)d_out;

    zero_gram_kernel<<<16, 256, 0, stream>>>(gram);
    gram_wmma_kernel<<<NBLOCKS, BLOCKT, 0, stream>>>(x, gram);
    finalize_kernel<<<1, 256, 0, stream>>>(gram, out);
}